// _SpatialAttentionMemory_26371099197795
// MI455X (gfx1250) — compile-verified
//
#include <hip/hip_runtime.h>

#define B_    4
#define N_    2048
#define K_    3072          // C*P
#define E_    128
#define ROWS  8192          // B*N
#define ROWS2 16384         // both embeds stacked
#define STRIPS 512          // ROWS2 / 32   (32-row strips)

typedef __attribute__((ext_vector_type(2))) float v2f;
typedef __attribute__((ext_vector_type(8))) float v8f;

// workspace layout (in floats)
#define Y_OFF   ((size_t)0)
#define Y_SZ    ((size_t)ROWS2 * E_)          // 2,097,152
#define PART_OFF (Y_OFF + Y_SZ)
#define PART_SZ ((size_t)STRIPS * 256)        // 131,072
#define STAT_OFF (PART_OFF + PART_SZ)
#define STAT_SZ ((size_t)512)
#define WS_FLOATS (STAT_OFF + STAT_SZ)

// ---------------- Kernel 1: fused dual-embed GEMM via V_WMMA_F32_16X16X4_F32
// Y[16384,128] = [sig; mem] (row-major [*,3072]) x conv_w^T ; per-strip stats partials.
// Block = 8 waves over a 32-row strip: wave = (row-group 0/1) x (e col-pair 0..3).
// Each wave: 2 accumulators (e0, e0+16) -> per K-step 1 A-load + 2 B-loads feed 2 WMMAs.
__global__ __launch_bounds__(256) void k_embed_gemm(
    const float* __restrict__ sig, const float* __restrict__ mem,
    const float* __restrict__ W, const float* __restrict__ cb,
    float* __restrict__ Y, float* __restrict__ part) {
  const int tid  = threadIdx.x;
  const int wv   = tid >> 5;
  const int lane = tid & 31;
  const int h    = lane >> 4;          // half-wave
  const int l16  = lane & 15;
  const int rg   = wv >> 2;            // row-group 0/1
  const int cp   = wv & 3;             // col pair -> e0 = cp*32
  const int e0   = cp << 5;
  const int s    = blockIdx.x;         // 32-row strip 0..511
  const size_t row = (size_t)s * 32 + rg * 16 + l16;
  const float* xrow = (s < (STRIPS / 2)) ? (sig + row * K_)
                                         : (mem + (row - (size_t)ROWS) * K_);
  const float* wrowA = W + (size_t)(e0 + l16) * K_;
  const float* wrowB = W + (size_t)(e0 + 16 + l16) * K_;
  const int kh = h * 2;

  v8f accA = {}, accB = {};
#pragma unroll 8
  for (int k = 0; k < K_; k += 4) {
    v2f a  = *(const v2f*)(xrow + k + kh);
    v2f bA = *(const v2f*)(wrowA + k + kh);
    v2f bB = *(const v2f*)(wrowB + k + kh);
    accA = __builtin_amdgcn_wmma_f32_16x16x4_f32(false, a, false, bA,
                                                 (short)0, accA, false, false);
    accB = __builtin_amdgcn_wmma_f32_16x16x4_f32(false, a, false, bB,
                                                 (short)0, accB, false, false);
  }

  const float cbA = cb[e0 + l16];
  const float cbB = cb[e0 + 16 + l16];
  float s1A = 0.f, s2A = 0.f, s1B = 0.f, s2B = 0.f;
#pragma unroll
  for (int v = 0; v < 8; ++v) {
    const size_t rowv = (size_t)s * 32 + rg * 16 + v + 8 * h;
    float yA = accA[v] + cbA;          // C[rowv, e0 + l16]
    float yB = accB[v] + cbB;          // C[rowv, e0 + 16 + l16]
    Y[rowv * E_ + (e0 + l16)]      = yA;
    Y[rowv * E_ + (e0 + 16 + l16)] = yB;
    s1A += yA; s2A += yA * yA;
    s1B += yB; s2B += yB * yB;
  }
  __shared__ float red[256][4];
  red[tid][0] = s1A; red[tid][1] = s2A;
  red[tid][2] = s1B; red[tid][3] = s2B;
  __syncthreads();
  {
    // thread t accumulates one (channel, stat) pair: chan = t>>1, c = t&1
    const int t    = tid;
    const int chan = t >> 1;
    const int c    = t & 1;
    const int cp2  = chan >> 5;          // which col-pair wave
    const int tl   = (chan >> 4) & 1;    // tile A(0) / B(1) within the pair
    const int l    = chan & 15;
    float sum = 0.f;
#pragma unroll
    for (int rg2 = 0; rg2 < 2; ++rg2)
#pragma unroll
      for (int hh = 0; hh < 2; ++hh)
        sum += red[((rg2 * 4 + cp2) << 5) + hh * 16 + l][tl * 2 + c];
    part[(size_t)s * 256 + t] = sum;     // t == chan*2 + c
  }
}

// ---------------- Kernel 2: reduce partials -> per-channel scale/shift (per half)
__global__ __launch_bounds__(256) void k_stats(
    const float* __restrict__ part, const float* __restrict__ gamma,
    const float* __restrict__ beta, float* __restrict__ stat) {
  const int bh = blockIdx.x;           // half: 0 = sig embed, 1 = mem embed
  const int t  = threadIdx.x;          // t = chan*2 + c
  const float* p = part + (size_t)bh * (STRIPS / 2) * 256 + t;
  float acc = 0.f;
  for (int k = 0; k < (STRIPS / 2); ++k) acc += p[(size_t)k * 256];
  __shared__ float sm[256];
  sm[t] = acc;
  __syncthreads();
  if (t < 128) {
    const float inv = 1.f / 8192.f;
    float mu  = sm[2 * t] * inv;
    float m2  = sm[2 * t + 1] * inv;
    float var = fmaxf(m2 - mu * mu, 0.f);
    float sc  = gamma[t] * rsqrtf(var + 1e-5f);
    float sh  = beta[t] - mu * sc;
    stat[((size_t)bh * 128 + t) * 2 + 0] = sc;
    stat[((size_t)bh * 128 + t) * 2 + 1] = sh;
  }
}

// ---------------- Kernel 3: in-place BN affine + LeakyReLU(0.1)
__global__ __launch_bounds__(256) void k_bn_lrelu(float* __restrict__ Y,
                                                  const float* __restrict__ stat) {
  const size_t idx = (size_t)blockIdx.x * 256 + threadIdx.x;  // < 2^21
  const int e    = (int)(idx & (E_ - 1));
  const int half = (int)(idx >> 20);                          // 8192*128 = 2^20
  const float sc = stat[((size_t)half * 128 + e) * 2 + 0];
  const float sh = stat[((size_t)half * 128 + e) * 2 + 1];
  float v = Y[idx] * sc + sh;
  Y[idx] = (v >= 0.f) ? v : 0.1f * v;
}

// ---------------- Kernel 4: banded sim + softmax; writes full attn row (zeros + band)
__global__ __launch_bounds__(128) void k_attn(const float* __restrict__ Y,
                                              float* __restrict__ attn) {
  const int r    = blockIdx.x;          // b*2048 + i
  const int i    = r & (N_ - 1);
  const int tid  = threadIdx.x;
  const int wv   = tid >> 5;
  const int lane = tid & 31;
  __shared__ float4 ex4[32];
  __shared__ float  sv[32];
  if (tid < 32) ex4[tid] = ((const float4*)(Y + (size_t)r * E_))[tid];

  // zero-fill the whole 2048-wide attn row
  float4 z4 = {0.f, 0.f, 0.f, 0.f};
  float4* row4 = (float4*)(attn + (size_t)r * N_);
#pragma unroll
  for (int q = 0; q < 4; ++q) row4[tid + 128 * q] = z4;
  __syncthreads();

  const int jlo = (i - 15 > 0) ? (i - 15) : 0;
  const int jhi = (i + 15 < N_ - 1) ? (i + 15) : (N_ - 1);
  const int nb  = jhi - jlo + 1;
  const int tb  = (r & ~(N_ - 1));      // b*2048

  for (int jj = wv; jj < nb; jj += 4) {
    const float4 et = ((const float4*)(Y + ((size_t)ROWS + tb + jlo + jj) * E_))[lane];
    const float4 xv = ex4[lane];
    float d = et.x * xv.x + et.y * xv.y + et.z * xv.z + et.w * xv.w;
    for (int off = 16; off >= 1; off >>= 1) d += __shfl_xor(d, off, 32);
    if (lane == 0) sv[jj] = d;
  }
  __syncthreads();

  if (wv == 0) {
    float v = (lane < nb) ? sv[lane] : -3.4e38f;
    float m = v;
    for (int off = 16; off >= 1; off >>= 1) m = fmaxf(m, __shfl_xor(m, off, 32));
    float e = (lane < nb) ? __expf(v - m) : 0.f;
    float su = e;
    for (int off = 16; off >= 1; off >>= 1) su += __shfl_xor(su, off, 32);
    if (lane < nb) attn[(size_t)r * N_ + jlo + lane] = e / su;
  }
}

// ---------------- Kernel 5: mem_w (banded weighted sum) + blend
__global__ __launch_bounds__(256) void k_memw(const float* __restrict__ attn,
                                              const float* __restrict__ sig,
                                              const float* __restrict__ mem,
                                              float* __restrict__ outn) {
  const int r   = blockIdx.x;           // b*2048 + i
  const int i   = r & (N_ - 1);
  const int tid = threadIdx.x;
  const int jlo = (i - 15 > 0) ? (i - 15) : 0;
  const int jhi = (i + 15 < N_ - 1) ? (i + 15) : (N_ - 1);
  const int nb  = jhi - jlo + 1;
  __shared__ float wsh[32];
  if (tid < nb) wsh[tid] = attn[(size_t)r * N_ + jlo + tid];
  __syncthreads();

  const int rb = (r & ~(N_ - 1)) + jlo; // first banded mem row (global row idx)
  float4 a0 = {0, 0, 0, 0}, a1 = {0, 0, 0, 0}, a2 = {0, 0, 0, 0};
  for (int j = 0; j < nb; ++j) {
    const float wj = wsh[j];
    const float4* m4 = (const float4*)(mem + (size_t)(rb + j) * K_);
    const float4 v0 = m4[tid], v1 = m4[tid + 256], v2 = m4[tid + 512];
    a0.x += wj * v0.x; a0.y += wj * v0.y; a0.z += wj * v0.z; a0.w += wj * v0.w;
    a1.x += wj * v1.x; a1.y += wj * v1.y; a1.z += wj * v1.z; a1.w += wj * v1.w;
    a2.x += wj * v2.x; a2.y += wj * v2.y; a2.z += wj * v2.z; a2.w += wj * v2.w;
  }
  const float4* s4 = (const float4*)(sig + (size_t)r * K_);
  float4* o4 = (float4*)(outn + (size_t)r * K_);
  const float4 s0 = s4[tid], s1 = s4[tid + 256], s2 = s4[tid + 512];
  float4 o0, o1, o2;
  o0.x = 0.5f * s0.x + 0.5f * a0.x; o0.y = 0.5f * s0.y + 0.5f * a0.y;
  o0.z = 0.5f * s0.z + 0.5f * a0.z; o0.w = 0.5f * s0.w + 0.5f * a0.w;
  o1.x = 0.5f * s1.x + 0.5f * a1.x; o1.y = 0.5f * s1.y + 0.5f * a1.y;
  o1.z = 0.5f * s1.z + 0.5f * a1.z; o1.w = 0.5f * s1.w + 0.5f * a1.w;
  o2.x = 0.5f * s2.x + 0.5f * a2.x; o2.y = 0.5f * s2.y + 0.5f * a2.y;
  o2.z = 0.5f * s2.z + 0.5f * a2.z; o2.w = 0.5f * s2.w + 0.5f * a2.w;
  o4[tid] = o0; o4[tid + 256] = o1; o4[tid + 512] = o2;
}

extern "C" void kernel_launch(void* const* d_in, const int* in_sizes, int n_in,
                              void* d_out, int out_size, void* d_ws, size_t ws_size,
                              hipStream_t stream) {
  (void)in_sizes; (void)n_in; (void)out_size;
  const float* sig   = (const float*)d_in[0];
  const float* mem   = (const float*)d_in[1];
  const float* W     = (const float*)d_in[2];
  const float* cb    = (const float*)d_in[3];
  const float* gamma = (const float*)d_in[4];
  const float* beta  = (const float*)d_in[5];

  float* out_new  = (float*)d_out;                          // [4,2048,64,48]
  float* out_attn = out_new + (size_t)B_ * N_ * K_;         // [4,2048,2048]

  // scratch: prefer d_ws; fall back to the new_mem output region (only written by k_memw, last)
  float* scratch = (ws_size >= WS_FLOATS * sizeof(float)) ? (float*)d_ws : out_new;
  float* Yb   = scratch + Y_OFF;
  float* part = scratch + PART_OFF;
  float* stat = scratch + STAT_OFF;

  k_embed_gemm<<<STRIPS, 256, 0, stream>>>(sig, mem, W, cb, Yb, part);
  k_stats<<<2, 256, 0, stream>>>(part, gamma, beta, stat);
  k_bn_lrelu<<<(int)((Y_SZ) / 256), 256, 0, stream>>>(Yb, stat);
  k_attn<<<ROWS, 128, 0, stream>>>(Yb, out_attn);
  k_memw<<<ROWS, 256, 0, stream>>>(out_attn, sig, mem, out_new);
}